// TripletLoss_14147622273110
// MI455X (gfx1250) — compile-verified
//
#include <hip/hip_runtime.h>
#include <stdint.h>

#define TL_D 256               // feature dim (floats per row)
#define TL_WAVES 8             // waves per block (wave32 -> 256 threads)
#define TL_STAGES 8            // pipelined stages per block
#define TL_ROWS_PER_BLOCK (TL_WAVES * TL_STAGES)   // 64 rows / block
#define TL_MARGIN 0.2f

// ---------------- CDNA5 async global->LDS path (ASYNCcnt) -------------------
// Per-lane 16B copy global memory -> LDS; whole wave copies 512B per issue.
// GV addressing: 64-bit address in VGPR pair, LDS byte offset in a VGPR.
__device__ __forceinline__ void tl_async_b128(uint32_t lds_addr, const void* gaddr) {
  asm volatile("global_load_async_to_lds_b128 %0, %1, off"
               :: "v"(lds_addr), "v"(gaddr)
               : "memory");
}
__device__ __forceinline__ void tl_wait_async6() { asm volatile("s_wait_asynccnt 6" ::: "memory"); }
__device__ __forceinline__ void tl_wait_async0() { asm volatile("s_wait_asynccnt 0" ::: "memory"); }
__device__ __forceinline__ void tl_wait_ds0()    { asm volatile("s_wait_dscnt 0"    ::: "memory"); }

// Per-row math: each lane holds 8 elements of a/p/n; reduce the 5 dot products
// across the 32-lane wave and return this row's clamped triplet loss (uniform
// across lanes after the xor-reduction).
__device__ __forceinline__ float tl_row_loss(float4 a0, float4 a1,
                                             float4 p0, float4 p1,
                                             float4 n0, float4 n1) {
  float av[8] = {a0.x, a0.y, a0.z, a0.w, a1.x, a1.y, a1.z, a1.w};
  float pv[8] = {p0.x, p0.y, p0.z, p0.w, p1.x, p1.y, p1.z, p1.w};
  float nv[8] = {n0.x, n0.y, n0.z, n0.w, n1.x, n1.y, n1.z, n1.w};
  float aa = 0.f, ap = 0.f, pp = 0.f, an = 0.f, nn = 0.f;
#pragma unroll
  for (int i = 0; i < 8; ++i) {
    aa = fmaf(av[i], av[i], aa);
    ap = fmaf(av[i], pv[i], ap);
    pp = fmaf(pv[i], pv[i], pp);
    an = fmaf(av[i], nv[i], an);
    nn = fmaf(nv[i], nv[i], nn);
  }
#pragma unroll
  for (int off = 16; off > 0; off >>= 1) {
    aa += __shfl_xor(aa, off, 32);
    ap += __shfl_xor(ap, off, 32);
    pp += __shfl_xor(pp, off, 32);
    an += __shfl_xor(an, off, 32);
    nn += __shfl_xor(nn, off, 32);
  }
  float dp = 1.0f - ap / sqrtf(pp * aa);
  float dn = 1.0f - an / sqrtf(nn * aa);
  float l  = dp - dn + TL_MARGIN;
  return l > 0.0f ? l : 0.0f;
}

__global__ __launch_bounds__(256) void tl_partial_kernel(
    const float* __restrict__ A, const float* __restrict__ P,
    const float* __restrict__ Ng, float* __restrict__ partial, int nRows) {
  // Double-buffered staging: [buf 2][arr 3][row 8][elem 256] floats = 48 KB.
  __shared__ __align__(16) float smem[2 * 3 * TL_WAVES * TL_D];
  __shared__ float wsum[TL_WAVES];

  const int tid     = (int)threadIdx.x;
  const int wave    = tid >> 5;
  const int lane    = tid & 31;
  const int rowBase = (int)blockIdx.x * TL_ROWS_PER_BLOCK;

  const int   arrS    = TL_WAVES * TL_D;           // floats per array slab
  const float* arrs[3] = {A, P, Ng};

  float lossAcc = 0.0f;

  if (rowBase + TL_ROWS_PER_BLOCK <= nRows) {
    // ---- fast path: async double-buffered pipeline -------------------------
    const uint32_t ldsBase   = (uint32_t)(uintptr_t)(&smem[0]);
    const uint32_t arrStride = (uint32_t)(arrS * 4);   // bytes per array slab

    auto issueStage = [&](int stage, int buf) {
      const int row = rowBase + stage * TL_WAVES + wave;
      const size_t g = (size_t)row * TL_D + (size_t)(lane * 4);   // float index
      const uint32_t lbase =
          ldsBase + (uint32_t)(((buf * 3) * TL_WAVES + wave) * TL_D + lane * 4) * 4u;
#pragma unroll
      for (int a = 0; a < 3; ++a) {
        tl_async_b128(lbase + (uint32_t)a * arrStride,        arrs[a] + g);
        tl_async_b128(lbase + (uint32_t)a * arrStride + 512u, arrs[a] + g + 128);
      }
    };

    auto computeStage = [&](int buf) {
      const int sbase = ((buf * 3) * TL_WAVES + wave) * TL_D + lane * 4;
      float4 a0 = *(const float4*)&smem[sbase];
      float4 a1 = *(const float4*)&smem[sbase + 128];
      float4 p0 = *(const float4*)&smem[sbase + arrS];
      float4 p1 = *(const float4*)&smem[sbase + arrS + 128];
      float4 n0 = *(const float4*)&smem[sbase + 2 * arrS];
      float4 n1 = *(const float4*)&smem[sbase + 2 * arrS + 128];
      lossAcc += tl_row_loss(a0, a1, p0, p1, n0, n1);
    };

    issueStage(0, 0);
#pragma unroll
    for (int s = 0; s < TL_STAGES; ++s) {
      if (s + 1 < TL_STAGES) {
        tl_wait_ds0();                       // our ds reads of the recycled buffer are done
        issueStage(s + 1, (s + 1) & 1);      // prefetch next stage (6 async ops in flight +6)
        tl_wait_async6();                    // oldest 6 (stage s) have landed in LDS
      } else {
        tl_wait_async0();                    // drain: last stage landed
      }
      computeStage(s & 1);
    }
  } else {
    // ---- tail path: plain coalesced global b128 loads ----------------------
    for (int s = 0; s < TL_STAGES; ++s) {
      const int row = rowBase + s * TL_WAVES + wave;
      if (row < nRows) {
        const size_t g = (size_t)row * TL_D + (size_t)(lane * 4);
        float4 a0 = *(const float4*)(A + g);
        float4 a1 = *(const float4*)(A + g + 128);
        float4 p0 = *(const float4*)(P + g);
        float4 p1 = *(const float4*)(P + g + 128);
        float4 n0 = *(const float4*)(Ng + g);
        float4 n1 = *(const float4*)(Ng + g + 128);
        lossAcc += tl_row_loss(a0, a1, p0, p1, n0, n1);
      }
    }
  }

  if (lane == 0) wsum[wave] = lossAcc;
  __syncthreads();
  if (tid == 0) {
    float s = 0.0f;
#pragma unroll
    for (int i = 0; i < TL_WAVES; ++i) s += wsum[i];
    partial[blockIdx.x] = s;
  }
}

// Deterministic final reduction: single block, fixed-order tree.
__global__ __launch_bounds__(256) void tl_reduce_kernel(const float* __restrict__ partial,
                                                        int nPart, float* __restrict__ out) {
  __shared__ float sb[256];
  float s = 0.0f;
  for (int i = (int)threadIdx.x; i < nPart; i += 256) s += partial[i];
  sb[threadIdx.x] = s;
  __syncthreads();
#pragma unroll
  for (int off = 128; off > 0; off >>= 1) {
    if ((int)threadIdx.x < off) sb[threadIdx.x] += sb[threadIdx.x + off];
    __syncthreads();
  }
  if (threadIdx.x == 0) out[0] = sb[0];
}

extern "C" void kernel_launch(void* const* d_in, const int* in_sizes, int n_in,
                              void* d_out, int out_size, void* d_ws, size_t ws_size,
                              hipStream_t stream) {
  (void)n_in; (void)out_size; (void)ws_size;
  const float* A  = (const float*)d_in[0];
  const float* P  = (const float*)d_in[1];
  const float* Ng = (const float*)d_in[2];
  const int nRows   = in_sizes[0] / TL_D;
  const int nBlocks = (nRows + TL_ROWS_PER_BLOCK - 1) / TL_ROWS_PER_BLOCK;
  float* partial = (float*)d_ws;

  tl_partial_kernel<<<nBlocks, 256, 0, stream>>>(A, P, Ng, partial, nRows);
  tl_reduce_kernel<<<1, 256, 0, stream>>>(partial, nBlocks, (float*)d_out);
}